// HiFiCodecQuantizer_42210938585804
// MI455X (gfx1250) — compile-verified
//
#include <hip/hip_runtime.h>

typedef float v2f __attribute__((ext_vector_type(2)));
typedef float v8f __attribute__((ext_vector_type(8)));

#define DIMC 512
#define TT   4096
#define BB   8
#define NG   8
#define NC   1024
#define SUBD 64
#define TTILE 32
#define QN   (BB*DIMC*TT)          /* 16777216 quantized elems   */
#define NN   (BB*TT)               /* 32768 vectors              */
#define NBLK (NN/TTILE)            /* 1024 workgroups            */
#define LOSS_SCALE (1.25f / (2.0f * 16777216.0f))

// ---------------------------------------------------------------------------
// Prep: codebook squared norms e2[st][g][k] into ws[0 .. 16384)
// ---------------------------------------------------------------------------
__global__ __launch_bounds__(256) void rvq_prep(const float* __restrict__ cb1,
                                                const float* __restrict__ cb2,
                                                float* __restrict__ ws) {
  int t = blockIdx.x * 256 + threadIdx.x;       // 0..16383
  const float* cb = (t < NG * NC) ? cb1 : cb2;
  int r = t & (NG * NC - 1);
  const float* row = cb + (size_t)r * SUBD;
  float s = 0.f;
#pragma unroll 8
  for (int i = 0; i < SUBD; ++i) s += row[i] * row[i];
  ws[t] = s;
}

// ---------------------------------------------------------------------------
// Main: residual VQ, 2 stages, WMMA fp32 distance GEMM + fused argmin
// ---------------------------------------------------------------------------
__global__ __launch_bounds__(256) void rvq_main(const float* __restrict__ x,
                                                const float* __restrict__ cb1,
                                                const float* __restrict__ cb2,
                                                const float* __restrict__ e2all,
                                                float* __restrict__ out,
                                                float* __restrict__ partial) {
  extern __shared__ float lds[];
  float* tile = lds;                               // [512][32] x tile -> q tile
  int*   ids  = (int*)(lds + DIMC * TTILE);        // [8][32] winning indices
  float* warr = (float*)(ids + NG * TTILE);        // [8] per-wave loss partials

  const int blk = blockIdx.x;
  const int b   = blk >> 7;                        // 128 tiles per batch elem
  const int t0  = (blk & 127) * TTILE;
  const int tid = threadIdx.x;

  // Cooperative coalesced load of x[b][:, t0:t0+32] into LDS (float4 rows)
  const float* xb = x + ((size_t)b * DIMC) * TT + t0;
  for (int i = tid; i < DIMC * TTILE / 4; i += 256) {
    int c = i >> 3, tq = (i & 7) << 2;
    *(float4*)(tile + c * TTILE + tq) = *(const float4*)(xb + (size_t)c * TT + tq);
  }
  __syncthreads();

  const int w    = tid >> 5;                       // wave id == group id
  const int lane = tid & 31;
  const int half = lane >> 4;                      // which 16-lane half
  const int m    = lane & 15;
  const int kb   = half << 1;                      // K offset {0,2} per half
  const int cbase = w * SUBD;

  // Residual in WMMA 16x4 fp32 A-layout: lane holds row m of each M-tile h,
  // VGPR pair s = K values {4s+kb, 4s+kb+1}.
  v2f rA[2][16];
#pragma unroll
  for (int h = 0; h < 2; ++h)
#pragma unroll
    for (int s = 0; s < 16; ++s) {
      int d = 4 * s + kb;
      rA[h][s].x = tile[(cbase + d)     * TTILE + h * 16 + m];
      rA[h][s].y = tile[(cbase + d + 1) * TTILE + h * 16 + m];
    }

  float lsum = 0.f;

  for (int st = 0; st < 2; ++st) {
    const float* cb  = (st ? cb2 : cb1) + (size_t)w * NC * SUBD;
    const float* e2g = e2all + (size_t)(st * NG + w) * NC;

    float mind[2][8];
    int   midx[2][8];
#pragma unroll
    for (int j = 0; j < 8; ++j) {
      mind[0][j] = 3.0e38f; mind[1][j] = 3.0e38f;
      midx[0][j] = 0;       midx[1][j] = 0;
    }

    for (int kt = 0; kt < 64; ++kt) {
      const int col = kt * 16 + m;                 // this lane's code column
      const float* crow = cb + (size_t)col * SUBD + kb;
      v2f bt[16];                                  // B operand: code row slices
#pragma unroll
      for (int s = 0; s < 16; ++s) bt[s] = *(const v2f*)(crow + 4 * s);

      // 4 independent accumulator chains -> dependent-WMMA distance of 4
      // instructions, hiding the matrix-pipe result latency.
      v8f acc0a = {};
      v8f acc0b = {};
      v8f acc1a = {};
      v8f acc1b = {};
#pragma unroll
      for (int s = 0; s < 16; s += 2) {
        acc0a = __builtin_amdgcn_wmma_f32_16x16x4_f32(false, rA[0][s],     false, bt[s],
                                                      (short)0, acc0a, false, false);
        acc1a = __builtin_amdgcn_wmma_f32_16x16x4_f32(false, rA[1][s],     false, bt[s],
                                                      (short)0, acc1a, false, false);
        acc0b = __builtin_amdgcn_wmma_f32_16x16x4_f32(false, rA[0][s + 1], false, bt[s + 1],
                                                      (short)0, acc0b, false, false);
        acc1b = __builtin_amdgcn_wmma_f32_16x16x4_f32(false, rA[1][s + 1], false, bt[s + 1],
                                                      (short)0, acc1b, false, false);
      }
      v8f acc0 = acc0a + acc0b;
      v8f acc1 = acc1a + acc1b;

      const float e2v = e2g[col];
#pragma unroll
      for (int j = 0; j < 8; ++j) {
        float d0 = fmaf(-2.f, acc0[j], e2v);       // dist - x2 (argmin invariant)
        float d1 = fmaf(-2.f, acc1[j], e2v);
        if (d0 < mind[0][j]) { mind[0][j] = d0; midx[0][j] = col; }
        if (d1 < mind[1][j]) { mind[1][j] = d1; midx[1][j] = col; }
      }
    }

    // argmin reduction across the 16 lanes of each half (rows stay separate)
#pragma unroll
    for (int off = 1; off <= 8; off <<= 1) {
#pragma unroll
      for (int h = 0; h < 2; ++h)
#pragma unroll
        for (int j = 0; j < 8; ++j) {
          float od = __shfl_xor(mind[h][j], off, 32);
          int   oi = __shfl_xor(midx[h][j], off, 32);
          if (od < mind[h][j] || (od == mind[h][j] && oi < midx[h][j])) {
            mind[h][j] = od; midx[h][j] = oi;
          }
        }
    }

    // lanes 0 (rows 0-7) and 16 (rows 8-15) publish winners for both M-tiles
    if (m == 0) {
#pragma unroll
      for (int h = 0; h < 2; ++h)
#pragma unroll
        for (int j = 0; j < 8; ++j)
          ids[w * TTILE + h * 16 + half * 8 + j] = midx[h][j];
    }
    __syncthreads();

    // coalesced index output: out_idx[(st*8+g)*32768 + b*4096 + t0 + lane]
    out[QN + 1 + (size_t)(st * NG + w) * NN + (size_t)b * TT + t0 + lane] =
        (float)ids[w * TTILE + lane];

    // gather z, accumulate loss with pre-update residual, update residual,
    // accumulate quantized = z1 + z2 in the LDS tile (exact ref arithmetic)
#pragma unroll
    for (int h = 0; h < 2; ++h) {
      const int im = ids[w * TTILE + h * 16 + m];
      const float* zrow = cb + (size_t)im * SUBD + kb;
#pragma unroll
      for (int s = 0; s < 16; ++s) {
        v2f z = *(const v2f*)(zrow + 4 * s);
        float dx = z.x - rA[h][s].x, dy = z.y - rA[h][s].y;
        lsum += dx * dx + dy * dy;
        rA[h][s].x -= z.x;
        rA[h][s].y -= z.y;
        int d = 4 * s + kb;
        float* q0 = &tile[(cbase + d)     * TTILE + h * 16 + m];
        float* q1 = &tile[(cbase + d + 1) * TTILE + h * 16 + m];
        if (st == 0) { *q0 = z.x;  *q1 = z.y;  }
        else         { *q0 += z.x; *q1 += z.y; }
      }
    }
    __syncthreads();
  }

  // deterministic block loss partial: wave shuffle-reduce -> LDS -> thread 0
#pragma unroll
  for (int off = 16; off >= 1; off >>= 1) lsum += __shfl_xor(lsum, off, 32);
  if (lane == 0) warr[w] = lsum;
  __syncthreads();
  if (tid == 0) {
    float s = 0.f;
#pragma unroll
    for (int i = 0; i < NG; ++i) s += warr[i];
    partial[blk] = s;
  }

  // coalesced store of quantized tile
  float* qout = out + ((size_t)b * DIMC) * TT + t0;
  for (int i = tid; i < DIMC * TTILE / 4; i += 256) {
    int c = i >> 3, tq = (i & 7) << 2;
    *(float4*)(qout + (size_t)c * TT + tq) = *(const float4*)(tile + c * TTILE + tq);
  }
}

// ---------------------------------------------------------------------------
// Finalize: deterministic sum of 1024 block partials -> scalar loss
// ---------------------------------------------------------------------------
__global__ __launch_bounds__(256) void rvq_loss(const float* __restrict__ partial,
                                                float* __restrict__ out) {
  __shared__ float sm[256];
  float s = 0.f;
  for (int i = threadIdx.x; i < NBLK; i += 256) s += partial[i];
  sm[threadIdx.x] = s;
  __syncthreads();
  for (int o = 128; o > 0; o >>= 1) {
    if (threadIdx.x < o) sm[threadIdx.x] += sm[threadIdx.x + o];
    __syncthreads();
  }
  if (threadIdx.x == 0) out[QN] = sm[0] * LOSS_SCALE;
}

// ---------------------------------------------------------------------------
extern "C" void kernel_launch(void* const* d_in, const int* in_sizes, int n_in,
                              void* d_out, int out_size, void* d_ws, size_t ws_size,
                              hipStream_t stream) {
  const float* x   = (const float*)d_in[0];
  const float* cb1 = (const float*)d_in[1];
  const float* cb2 = (const float*)d_in[2];
  float* out = (float*)d_out;
  float* ws  = (float*)d_ws;            // [0,16384): e2, [16384,17408): partials

  rvq_prep<<<64, 256, 0, stream>>>(cb1, cb2, ws);

  size_t ldsBytes = (size_t)(DIMC * TTILE) * 4 + NG * TTILE * 4 + NG * 4; // 66592 B
  rvq_main<<<NBLK, 256, ldsBytes, stream>>>(x, cb1, cb2, ws, out, ws + 2 * NG * NC);

  rvq_loss<<<1, 256, 0, stream>>>(ws + 2 * NG * NC, out);
}